// DistGATConv_19610820673602
// MI455X (gfx1250) — compile-verified
//
#include <hip/hip_runtime.h>
#include <hip/hip_bf16.h>

// ---------------- problem constants (match reference) ----------------
#define N_SRC     100000
#define N_DST     20000
#define N_EDGE    640000
#define IN_FEATS  256
#define NUM_HEADS 4
#define OUT_FEATS 32
#define HD        (NUM_HEADS * OUT_FEATS)   // 128
#define NEG_SLOPE 0.2f
#define MTILES    (N_SRC / 16)              // 6250 (exact)

typedef __attribute__((ext_vector_type(2))) float v2f;
typedef __attribute__((ext_vector_type(4))) float v4f;
typedef __attribute__((ext_vector_type(8))) float v8f;

// =====================================================================
// Kernel 1: h = x @ W via V_WMMA_F32_16X16X4_F32.
// Block = 256 threads = 8 waves. Wave w handles M-tile (blockIdx.x*8 + w),
// computing all 8 N-tiles (N = 128) with 8 v8f accumulators.
// W staged through LDS in two 64KB halves, K-PAIR INTERLEAVED:
//   sW[(k/2)*256 + n*2 + (k&1)] = W[k0+k][n]
// so each WMMA B operand (K=ka, K=ka+1 for one n) is a single aligned
// 8-byte contiguous ds_load_b64 straight into an even VGPR pair.
// =====================================================================
__global__ __launch_bounds__(256, 1)
void gat_gemm_wmma(const float* __restrict__ x, const float* __restrict__ W,
                   float* __restrict__ h) {
  __shared__ __align__(16) float sW[128 * HD];   // 64 KB

  const int wave  = threadIdx.x >> 5;
  const int lane  = threadIdx.x & 31;
  int mtile = blockIdx.x * 8 + wave;
  if (mtile >= MTILES) mtile = MTILES - 1;  // clamp: duplicate identical work

  const int row   = lane & 15;              // M (for A) / N (for B,C,D) index
  const int khalf = lane >> 4;              // 0 or 1 (K sub-pair / M high half)

  v8f acc[8];
#pragma unroll
  for (int nt = 0; nt < 8; ++nt) acc[nt] = (v8f){0.f,0.f,0.f,0.f,0.f,0.f,0.f,0.f};

  const float* xrow = x + (size_t)(mtile * 16 + row) * IN_FEATS;
  const v2f* sWp = (const v2f*)sW;          // float2-granular view of LDS

  for (int k0 = 0; k0 < IN_FEATS; k0 += 128) {
    // cooperative stage, global-coalesced: i = kk*256 + p*128 + n
    for (int i = threadIdx.x; i < 128 * HD; i += 256) {
      const int kk = i >> 8;
      const int p  = (i >> 7) & 1;
      const int n  = i & 127;
      sW[kk * 256 + n * 2 + p] = W[(size_t)(k0 + kk * 2 + p) * HD + n];
    }
    __syncthreads();

#pragma unroll 4
    for (int k = 0; k < 128; k += 4) {
      const int ka = k + khalf * 2;
      // A fp32 16x4: VGPR0 = K=(2*khalf), VGPR1 = K=(2*khalf+1), M = lane%16
      v2f a;
      a.x = xrow[k0 + ka];
      a.y = xrow[k0 + ka + 1];
      const int kp = (ka >> 1) * 128;       // float2 index of K-pair row
#pragma unroll
      for (int nt = 0; nt < 8; ++nt) {
        // B fp32 4x16: one b64 LDS load -> (K=ka, K=ka+1) for n = nt*16+row
        v2f b = sWp[kp + nt * 16 + row];
        acc[nt] = __builtin_amdgcn_wmma_f32_16x16x4_f32(
            /*neg_a=*/false, a, /*neg_b=*/false, b,
            /*c_mod=*/(short)0, acc[nt], /*reuse_a=*/false, /*reuse_b=*/false);
      }
    }
    __syncthreads();
  }

  // C/D layout: VGPR r -> M = r + 8*khalf, N = lane%16
  float* hp = h + (size_t)mtile * 16 * HD;
#pragma unroll
  for (int nt = 0; nt < 8; ++nt) {
#pragma unroll
    for (int r = 0; r < 8; ++r) {
      const int m = r + khalf * 8;
      hp[(size_t)m * HD + nt * 16 + row] = acc[nt][r];
    }
  }
}

// =====================================================================
// Kernel 2: per-node attention scores.
//   map == nullptr : el[n,h] = dot(h[n,h,:], attn[h,:])           (n < N_SRC)
//   map != nullptr : er[n,h] = dot(h[map[n],h,:], attn[h,:])      (n < N_DST)
// =====================================================================
__global__ __launch_bounds__(256)
void gat_node_score(const float* __restrict__ h, const float* __restrict__ attn,
                    const int* __restrict__ map, float* __restrict__ score,
                    int n_nodes) {
  const int t = blockIdx.x * blockDim.x + threadIdx.x;
  if (t >= n_nodes * NUM_HEADS) return;
  const int node = t >> 2;
  const int hh   = t & 3;
  const int src  = map ? map[node] : node;
  const float* hp = h + (size_t)src * HD + hh * OUT_FEATS;
  const float* ap = attn + hh * OUT_FEATS;
  float s = 0.f;
#pragma unroll
  for (int d = 0; d < OUT_FEATS; ++d) s += hp[d] * ap[d];
  score[t] = s;
}

// =====================================================================
// Kernel 3: one thread per EDGE, all 4 heads vectorized.
// e[edge,:] = leaky_relu(el[src,:] + er[dst,:]); atomic segment-sum.
// float4 loads of el/er, float4 store of e, one src/dst index read per edge.
// =====================================================================
__global__ __launch_bounds__(256)
void gat_edge_score(const int* __restrict__ src_idx, const int* __restrict__ dst_idx,
                    const float* __restrict__ el, const float* __restrict__ er,
                    float* __restrict__ e, float* __restrict__ sum_e) {
  const int edge = blockIdx.x * blockDim.x + threadIdx.x;
  if (edge >= N_EDGE) return;
  const int s = src_idx[edge];
  const int d = dst_idx[edge];
  const v4f vl = *(const v4f*)(el + (size_t)s * NUM_HEADS);
  const v4f vr = *(const v4f*)(er + (size_t)d * NUM_HEADS);
  v4f v;
#pragma unroll
  for (int hh = 0; hh < NUM_HEADS; ++hh) {
    float t = vl[hh] + vr[hh];
    v[hh] = (t >= 0.f) ? t : NEG_SLOPE * t;
  }
  *(v4f*)(e + (size_t)edge * NUM_HEADS) = v;
#pragma unroll
  for (int hh = 0; hh < NUM_HEADS; ++hh)
    atomicAdd(&sum_e[d * NUM_HEADS + hh], v[hh]);
}

// =====================================================================
// Kernel 4: out[dst,h,d] += (e/sum_e) * h[src,h,d].
// One thread per (edge, 4-feature chunk): 32 threads (one wave) cover an
// edge's 128 features -> wave-uniform edge index (scalarized score loads),
// one b128 gather of h per thread, 4 coalesced atomics.
// =====================================================================
__global__ __launch_bounds__(256)
void gat_scatter(const int* __restrict__ src_idx, const int* __restrict__ dst_idx,
                 const float* __restrict__ e, const float* __restrict__ sum_e,
                 const float* __restrict__ h, float* __restrict__ out) {
  const long long t = (long long)blockIdx.x * blockDim.x + threadIdx.x;
  if (t >= (long long)N_EDGE * 32) return;
  const int edge = (int)(t >> 5);
  const int q    = (int)(t & 31);        // 4-float chunk index: feat = q*4..q*4+3
  const int hh   = q >> 3;               // head = (q*4)/32
  const int s = src_idx[edge];
  const int d = dst_idx[edge];
  const float a = e[edge * NUM_HEADS + hh] / sum_e[d * NUM_HEADS + hh];
  const v4f hv = *(const v4f*)(h + (size_t)s * HD + q * 4);
  float* op = out + (size_t)d * HD + q * 4;
#pragma unroll
  for (int j = 0; j < 4; ++j)
    atomicAdd(&op[j], a * hv[j]);
}

// =====================================================================
extern "C" void kernel_launch(void* const* d_in, const int* in_sizes, int n_in,
                              void* d_out, int out_size, void* d_ws, size_t ws_size,
                              hipStream_t stream) {
  const float* x          = (const float*)d_in[0];
  const float* W          = (const float*)d_in[1];
  const float* attn_l     = (const float*)d_in[2];
  const float* attn_r     = (const float*)d_in[3];
  const int*   src_idx    = (const int*)d_in[4];
  const int*   dst_idx    = (const int*)d_in[5];
  const int*   dst_to_src = (const int*)d_in[6];
  float* out = (float*)d_out;

  // workspace layout (floats): h | el | er | sum_e | e   (~63.7 MB total)
  float* ws    = (float*)d_ws;
  float* h     = ws;                                  // N_SRC*128 = 12.8M
  float* el    = h  + (size_t)N_SRC * HD;             // 400k
  float* er    = el + (size_t)N_SRC * NUM_HEADS;      // 80k
  float* sum_e = er + (size_t)N_DST * NUM_HEADS;      // 80k
  float* e     = sum_e + (size_t)N_DST * NUM_HEADS;   // 2.56M

  // zero accumulation targets each call (graph-capture-safe stream memset)
  hipMemsetAsync(out,   0, (size_t)N_DST * HD * sizeof(float), stream);
  hipMemsetAsync(sum_e, 0, (size_t)N_DST * NUM_HEADS * sizeof(float), stream);

  // 1) h = x @ W  (WMMA). 6250 M-tiles, 8 waves/block -> 782 blocks.
  gat_gemm_wmma<<<dim3((MTILES + 7) / 8), dim3(256), 0, stream>>>(x, W, h);

  // 2) node scores
  gat_node_score<<<dim3((N_SRC * NUM_HEADS + 255) / 256), dim3(256), 0, stream>>>(
      h, attn_l, nullptr, el, N_SRC);
  gat_node_score<<<dim3((N_DST * NUM_HEADS + 255) / 256), dim3(256), 0, stream>>>(
      h, attn_r, dst_to_src, er, N_DST);

  // 3) edge scores + segment sum (one thread per edge)
  gat_edge_score<<<dim3((N_EDGE + 255) / 256), dim3(256), 0, stream>>>(
      src_idx, dst_idx, el, er, e, sum_e);

  // 4) normalized weighted scatter-add (one thread per edge-feature-quad)
  const long long total = (long long)N_EDGE * 32;   // 20.48M threads
  gat_scatter<<<dim3((unsigned)((total + 255) / 256)), dim3(256), 0, stream>>>(
      src_idx, dst_idx, e, sum_e, h, out);
}